// DistributedMPNN_77859167142091
// MI455X (gfx1250) — compile-verified
//
#include <hip/hip_runtime.h>
#include <math.h>

#define NN   8192
#define DEG  16
#define NE   (NN * DEG)
// var = (10^-19.9 * 5e6) / 10
#define VARF 6.2946270589708364e-15f

typedef float v2f __attribute__((ext_vector_type(2)));
typedef float v8f __attribute__((ext_vector_type(8)));

static __device__ __forceinline__ v8f wmma4(v2f a, v2f b, v8f c) {
  // D = A(16x4,f32) * B(4x16,f32) + C(16x16,f32)
  return __builtin_amdgcn_wmma_f32_16x16x4_f32(false, a, false, b, (short)0, c,
                                               false, false);
}

static __device__ __forceinline__ v8f relu8(v8f v) {
#pragma unroll
  for (int g = 0; g < 8; ++g) v[g] = fmaxf(v[g], 0.f);
  return v;
}

static __device__ __forceinline__ float sum8(v8f v) {
  float s = 0.f;
#pragma unroll
  for (int g = 0; g < 8; ++g) s += v[g];
  return s;
}

// One wave (32 lanes) per node. 8 nodes per 256-thread block.
__global__ __launch_bounds__(256) void mpnn_p_kernel(
    const float* __restrict__ x_in, const float* __restrict__ edge_attr,
    const float* __restrict__ Wm1, const float* __restrict__ bm1,
    const float* __restrict__ Wm2, const float* __restrict__ bm2,
    const float* __restrict__ Wu1, const float* __restrict__ bu1,
    const float* __restrict__ Wu2, const float* __restrict__ bu2,
    const float* __restrict__ Wh1, const float* __restrict__ bh1,
    const float* __restrict__ Wh2, const float* __restrict__ bh2,
    float* __restrict__ p_out) {
  const int lane = threadIdx.x & 31;
  const int wave = threadIdx.x >> 5;
  const int node = blockIdx.x * 8 + wave;
  const int r    = lane & 15;  // row / col index within a 16-tile
  const int hi   = lane >> 4;  // half-wave selector

  // wave-private staging tile for H1 (16x32, stride 34: even -> 8B aligned pairs)
  __shared__ float h1buf[8][16 * 34];
  float* h1 = h1buf[wave];

  // ---- preload weight B-fragments once, reuse across 3 iterations ----
  // Wm1 [10,32]: 3 K-steps x 2 N-tiles (K padded 10->12)
  const float* wm1p = Wm1 + (size_t)node * 320;
  v2f bf1[3][2];
#pragma unroll
  for (int k = 0; k < 3; ++k) {
    const int row0 = 4 * k + 2 * hi;
#pragma unroll
    for (int t = 0; t < 2; ++t) {
      const int col = 16 * t + r;
      bf1[k][t].x = (row0 < 10) ? wm1p[row0 * 32 + col] : 0.f;
      bf1[k][t].y = (row0 + 1 < 10) ? wm1p[(row0 + 1) * 32 + col] : 0.f;
    }
  }
  // Wm2 [32,32]: 8 K-steps x 2 N-tiles
  const float* wm2p = Wm2 + (size_t)node * 1024;
  v2f bf2[8][2];
#pragma unroll
  for (int k = 0; k < 8; ++k) {
    const int row0 = 4 * k + 2 * hi;
#pragma unroll
    for (int t = 0; t < 2; ++t) {
      const int col = 16 * t + r;
      bf2[k][t].x = wm2p[row0 * 32 + col];
      bf2[k][t].y = wm2p[(row0 + 1) * 32 + col];
    }
  }
  const float b1v0 = bm1[node * 32 + r];
  const float b1v1 = bm1[node * 32 + 16 + r];
  const float b2v0 = bm2[node * 32 + r];
  const float b2v1 = bm2[node * 32 + 16 + r];
  const float eaL  = edge_attr[node * DEG + r];  // edge r of this node

  float xv[9];
#pragma unroll
  for (int c = 0; c < 9; ++c) xv[c] = x_in[node * 9 + c];

  const float* wu1p = Wu1 + (size_t)node * 656;
  const float* wu2p = Wu2 + (size_t)node * 128;
  const float  bu1v = bu1[node * 16 + r];
  const float  bu2v = bu2[node * 8 + (lane & 7)];

  float comb = 0.f;

  for (int it = 0; it < 3; ++it) {
    // ---------- layer 1: M1 = relu([16x10] @ Wm1 + bm1) via 3 WMMA K-steps ----------
    v8f acc0 = {b1v0, b1v0, b1v0, b1v0, b1v0, b1v0, b1v0, b1v0};
    v8f acc1 = {b1v1, b1v1, b1v1, b1v1, b1v1, b1v1, b1v1, b1v1};
    v2f a;
    // k=0: cols (0,1) | (2,3)
    a.x = hi ? xv[2] : xv[0];
    a.y = hi ? xv[3] : xv[1];
    acc0 = wmma4(a, bf1[0][0], acc0);
    acc1 = wmma4(a, bf1[0][1], acc1);
    // k=1: cols (4,5) | (6,7)
    a.x = hi ? xv[6] : xv[4];
    a.y = hi ? xv[7] : xv[5];
    acc0 = wmma4(a, bf1[1][0], acc0);
    acc1 = wmma4(a, bf1[1][1], acc1);
    // k=2: cols (8,9) | (10,11)=pad
    a.x = hi ? 0.f : xv[8];
    a.y = hi ? 0.f : eaL;
    acc0 = wmma4(a, bf1[2][0], acc0);
    acc1 = wmma4(a, bf1[2][1], acc1);
    acc0 = relu8(acc0);
    acc1 = relu8(acc1);

    // stage H1 (D-layout -> row-major LDS) for layer-2 A-fragments
#pragma unroll
    for (int g = 0; g < 8; ++g) {
      const int row = g + 8 * hi;
      h1[row * 34 + r]      = acc0[g];
      h1[row * 34 + 16 + r] = acc1[g];
    }
    __syncthreads();

    // ---------- layer 2: M2 = relu(H1[16x32] @ Wm2 + bm2), 8 WMMA K-steps ----------
    v8f m0 = {b2v0, b2v0, b2v0, b2v0, b2v0, b2v0, b2v0, b2v0};
    v8f m1 = {b2v1, b2v1, b2v1, b2v1, b2v1, b2v1, b2v1, b2v1};
#pragma unroll
    for (int k = 0; k < 8; ++k) {
      const int c0 = 4 * k + 2 * hi;
      v2f af;
      af.x = h1[r * 34 + c0];
      af.y = h1[r * 34 + c0 + 1];
      m0 = wmma4(af, bf2[k][0], m0);
      m1 = wmma4(af, bf2[k][1], m1);
    }
    __syncthreads();  // protect LDS tile before next iteration overwrites
    m0 = relu8(m0);
    m1 = relu8(m1);

    // ---------- aggregate: column sums over the 16 edge-rows ----------
    float agg0 = sum8(m0);  // rows 0-7 (lanes 0-15) / 8-15 (lanes 16-31), col r
    float agg1 = sum8(m1);
    agg0 += __shfl_xor(agg0, 16, 32);  // lane L now holds agg[r]      (t=0)
    agg1 += __shfl_xor(agg1, 16, 32);  // lane L now holds agg[16 + r] (t=1)

    // ---------- update MLP: u=[x(9), agg(32)] -> 16 -> 8 ----------
    float a1 = bu1v;  // lane computes output j = r
#pragma unroll
    for (int i = 0; i < 9; ++i) a1 += xv[i] * wu1p[i * 16 + r];
#pragma unroll
    for (int i = 0; i < 16; ++i)
      a1 += __shfl(agg0, i, 32) * wu1p[(9 + i) * 16 + r];
#pragma unroll
    for (int i = 0; i < 16; ++i)
      a1 += __shfl(agg1, i, 32) * wu1p[(25 + i) * 16 + r];
    const float h2 = fmaxf(a1, 0.f);

    float a2 = bu2v;  // lane computes output o = lane&7
#pragma unroll
    for (int j = 0; j < 16; ++j)
      a2 += __shfl(h2, j, 32) * wu2p[j * 8 + (lane & 7)];
    comb = fmaxf(a2, 0.f);

    // ---------- x <- [x0, comb] ----------
#pragma unroll
    for (int c = 1; c < 9; ++c) xv[c] = __shfl(comb, c - 1, 32);
  }

  // ---------- head: p = sigmoid(relu(comb @ Wh1 + bh1) @ Wh2 + bh2) ----------
  float a3 = bh1[node * 16 + r];
  const float* wh1p = Wh1 + (size_t)node * 128;
#pragma unroll
  for (int i = 0; i < 8; ++i)
    a3 += __shfl(comb, i, 32) * wh1p[i * 16 + r];
  const float hh = fmaxf(a3, 0.f);
  float term = hh * Wh2[node * 16 + r];
  term += __shfl_xor(term, 8, 32);
  term += __shfl_xor(term, 4, 32);
  term += __shfl_xor(term, 2, 32);
  term += __shfl_xor(term, 1, 32);
  if (lane == 0) {
    const float z = term + bh2[node];
    p_out[node] = 1.f / (1.f + expf(-z));
  }
}

// One thread per node: interference gather + per-node rate, deterministic
// per-block tree reduction into partial[blockIdx].
__global__ __launch_bounds__(256) void rate_kernel(
    const int* __restrict__ edge_index, const float* __restrict__ H,
    const float* __restrict__ p, float* __restrict__ partial) {
  const int tid = threadIdx.x;
  const int i   = blockIdx.x * 256 + tid;
  const int* dstv = edge_index + NE;  // second row of [2, E]
  float interf = VARF;
#pragma unroll
  for (int k = 0; k < DEG; ++k) {
    const int d = dstv[i * DEG + k];
    interf += p[d] * H[(size_t)i * NN + d];
  }
  const float valid = p[i] * H[(size_t)i * NN + i];
  const float rate  = log2f(1.f + valid / interf);

  __shared__ float sbuf[256];
  sbuf[tid] = rate;
  __syncthreads();
  for (int s = 128; s > 0; s >>= 1) {
    if (tid < s) sbuf[tid] += sbuf[tid + s];
    __syncthreads();
  }
  if (tid == 0) partial[blockIdx.x] = sbuf[0];
}

__global__ void finalize_kernel(const float* __restrict__ partial,
                                float* __restrict__ out, int n) {
  if (threadIdx.x == 0 && blockIdx.x == 0) {
    float s = 0.f;
    for (int j = 0; j < n; ++j) s += partial[j];  // fixed order: deterministic
    out[0] = -s;
  }
}

extern "C" void kernel_launch(void* const* d_in, const int* in_sizes, int n_in,
                              void* d_out, int out_size, void* d_ws,
                              size_t ws_size, hipStream_t stream) {
  (void)in_sizes; (void)n_in; (void)out_size; (void)ws_size;
  const float* x         = (const float*)d_in[0];
  const float* edge_attr = (const float*)d_in[1];
  const int*   edge_idx  = (const int*)d_in[2];
  const float* H         = (const float*)d_in[3];
  const float* Wm1 = (const float*)d_in[4];
  const float* bm1 = (const float*)d_in[5];
  const float* Wm2 = (const float*)d_in[6];
  const float* bm2 = (const float*)d_in[7];
  const float* Wu1 = (const float*)d_in[8];
  const float* bu1 = (const float*)d_in[9];
  const float* Wu2 = (const float*)d_in[10];
  const float* bu2 = (const float*)d_in[11];
  const float* Wh1 = (const float*)d_in[12];
  const float* bh1 = (const float*)d_in[13];
  const float* Wh2 = (const float*)d_in[14];
  const float* bh2 = (const float*)d_in[15];

  float* p       = (float*)d_ws;   // NN floats
  float* partial = p + NN;         // NN/256 = 32 floats

  mpnn_p_kernel<<<NN / 8, 256, 0, stream>>>(x, edge_attr, Wm1, bm1, Wm2, bm2,
                                            Wu1, bu1, Wu2, bu2, Wh1, bh1, Wh2,
                                            bh2, p);
  rate_kernel<<<NN / 256, 256, 0, stream>>>(edge_idx, H, p, partial);
  finalize_kernel<<<1, 1, 0, stream>>>(partial, (float*)d_out, NN / 256);
}